// QGRU_86380382257535
// MI455X (gfx1250) — compile-verified
//
#include <hip/hip_runtime.h>
#include <math.h>

// MI455X / gfx1250, wave32. Quantum-GRU forward.
// Inputs (setup_inputs order): x(128,1024,128), W_in(4,256), b_in(4), W_x(4,128),
// b_x(4), W_h(4,128), b_h(4), W_out(128,4), b_out(128), vqc_w(4,1,4).
// Output: [hidden_seq (B,T,H) | h_last (B,H) | c_last zeros (B,H)] as f32.
//
// Latency-bound design: 1024 serial steps, 8 WGs x 8 wave32. Branch-free HW
// transcendentals; WMMA f32 16x16x4 for the (16x4)@(4x128) back-projections;
// 4-way split accumulators in the input projections to break FMA chains.

typedef __attribute__((ext_vector_type(2))) float v2f;
typedef __attribute__((ext_vector_type(8))) float v8f;

#define B_TOT 128
#define T_LEN 1024
#define F_DIM 128
#define H_DIM 128
#define TILE  16
#define XPAD  132   // row stride in floats: conflict-free for 16-row column walks

__device__ __forceinline__ float fast_rcp(float v) {
    return __builtin_amdgcn_rcpf(v);
}
__device__ __forceinline__ float fast_sigmoid(float v) {
    return fast_rcp(1.0f + __expf(-v));
}
__device__ __forceinline__ float fast_tanh(float v) {
    const float ax = fabsf(v);
    const float e  = __expf(-2.0f * ax);
    const float th = (1.0f - e) * fast_rcp(1.0f + e);
    return copysignf(th, v);
}

// 128-length dot product from LDS with 4 independent accumulator chains.
__device__ __forceinline__ float dot128(const float* __restrict__ a,
                                        const float* __restrict__ b) {
    const float4* a4 = reinterpret_cast<const float4*>(a);
    const float4* b4 = reinterpret_cast<const float4*>(b);
    float s0 = 0.f, s1 = 0.f, s2 = 0.f, s3 = 0.f;
    #pragma unroll 8
    for (int k = 0; k < 32; ++k) {
        const float4 u = a4[k];
        const float4 v = b4[k];
        s0 = fmaf(u.x, v.x, s0);
        s1 = fmaf(u.y, v.y, s1);
        s2 = fmaf(u.z, v.z, s2);
        s3 = fmaf(u.w, v.w, s3);
    }
    return (s0 + s1) + (s2 + s3);
}

__global__ __launch_bounds__(256) void qgru_kernel(
    const float* __restrict__ x,
    const float* __restrict__ W_in, const float* __restrict__ b_in,
    const float* __restrict__ W_x,  const float* __restrict__ b_x,
    const float* __restrict__ W_h,  const float* __restrict__ b_h,
    const float* __restrict__ W_out,const float* __restrict__ b_out,
    const float* __restrict__ vqc_w,
    float* __restrict__ out)
{
    __shared__ float sh_h[TILE][XPAD];        // hidden state tile (16 x 128)
    __shared__ float sh_x[2][TILE][XPAD];     // double-buffered x_t tile
    __shared__ float sh_Win[4][256];
    __shared__ float sh_Wx[4][128];
    __shared__ float sh_Wh[4][128];
    __shared__ float sh_bin[4], sh_bx[4], sh_bh[4];
    __shared__ float sh_vqcw[4][4];
    __shared__ float sh_part[TILE][8];        // y partial sums
    __shared__ float sh_axq[TILE][4];         // xq (with bias)
    __shared__ float sh_ahq[TILE][4];         // hq (with bias)
    __shared__ float sh_E[4][TILE][4];        // VQC <Z> per gate (WMMA A operands)

    const int tid = threadIdx.x;
    const int b0  = blockIdx.x * TILE;

    // ---- preload weights into LDS ----
    {
        float* p = &sh_Win[0][0];
        for (int k = tid; k < 4*256; k += 256) p[k] = W_in[k];
        p = &sh_Wx[0][0];
        for (int k = tid; k < 4*128; k += 256) p[k] = W_x[k];
        p = &sh_Wh[0][0];
        for (int k = tid; k < 4*128; k += 256) p[k] = W_h[k];
        if (tid < 4) { sh_bin[tid] = b_in[tid]; sh_bx[tid] = b_x[tid]; sh_bh[tid] = b_h[tid]; }
        if (tid < 16) sh_vqcw[tid >> 2][tid & 3] = vqc_w[tid];   // (4,1,4) flat
        p = &sh_h[0][0];
        for (int k = tid; k < TILE*XPAD; k += 256) p[k] = 0.0f;  // h0 = 0
    }

    // ---- per-lane WMMA constants (wave w owns H columns [16w,16w+16)) ----
    const int lane = tid & 31;
    const int wv   = tid >> 5;          // wave id 0..7
    const int half = lane >> 4;         // K-half selector
    const int nl   = lane & 15;         // row (A) / column (B,C)
    const int col  = wv * 16 + nl;      // global H column
    // B = W_out^T (4 x 16 slice): lane holds B[2*half..2*half+1][nl] = W_out[col][2*half..]
    v2f bmat;
    bmat.x = W_out[col * 4 + 2 * half + 0];
    bmat.y = W_out[col * 4 + 2 * half + 1];
    const float bn = b_out[col];        // C seeded with b_out broadcast over rows

    const int row = tid >> 4;           // 0..15 (batch row within tile)
    const int sub = tid & 15;           // task id within row

    // ---- load x_0 tile ----
    {
        const float4* src = reinterpret_cast<const float4*>(
            x + ((size_t)(b0 + row) * T_LEN + 0) * F_DIM + sub * 8);
        float4 r0 = src[0], r1 = src[1];
        float4* dst = reinterpret_cast<float4*>(&sh_x[0][row][sub * 8]);
        dst[0] = r0; dst[1] = r1;
    }
    __syncthreads();

    for (int t = 0; t < T_LEN; ++t) {
        const int cur = t & 1, nxt = cur ^ 1;

        // prefetch x_{t+1} into registers (hides HBM latency behind compute)
        float4 r0 = {0,0,0,0}, r1 = {0,0,0,0};
        if (t + 1 < T_LEN) {
            const float4* src = reinterpret_cast<const float4*>(
                x + ((size_t)(b0 + row) * T_LEN + (t + 1)) * F_DIM + sub * 8);
            r0 = src[0]; r1 = src[1];
        }

        // ---- Phase B: input projections (256 parallel 128-length dots) ----
        if (sub < 8) {              // y = W_in . [h ; x], split into h-half / x-half
            const int j = sub >> 1, hk = sub & 1;
            const float* wr = &sh_Win[j][hk * 128];
            const float* vv = hk ? &sh_x[cur][row][0] : &sh_h[row][0];
            sh_part[row][sub] = dot128(wr, vv);
        } else if (sub < 12) {      // xq = W_x . x + b_x
            const int j = sub - 8;
            sh_axq[row][j] = sh_bx[j] + dot128(&sh_Wx[j][0], &sh_x[cur][row][0]);
        } else {                    // hq = W_h . h + b_h
            const int j = sub - 12;
            sh_ahq[row][j] = sh_bh[j] + dot128(&sh_Wh[j][0], &sh_h[row][0]);
        }
        __syncthreads();

        // ---- Phase C: closed-form 4-qubit VQC (L=1) ----
        // RX(theta)RX(phi) = RX(theta+phi); conjugating Z_q through the CNOT ring
        // (0,1)(1,2)(2,3)(3,0) gives, with c_q = cos(angle_q + w_q):
        //   z0=c1c2c3, z1=c0c1, z2=c0c1c2, z3=c0c1c2c3
        if (tid < 64) {
            const int g = tid >> 4, i = tid & 15;   // gate, batch row
            float a0_, a1_, a2_, a3_;
            if (g < 2) {            // gates r,z use y
                a0_ = sh_part[i][0] + sh_part[i][1] + sh_bin[0];
                a1_ = sh_part[i][2] + sh_part[i][3] + sh_bin[1];
                a2_ = sh_part[i][4] + sh_part[i][5] + sh_bin[2];
                a3_ = sh_part[i][6] + sh_part[i][7] + sh_bin[3];
            } else if (g == 2) {    // n_x uses xq
                a0_ = sh_axq[i][0]; a1_ = sh_axq[i][1];
                a2_ = sh_axq[i][2]; a3_ = sh_axq[i][3];
            } else {                // n_h uses hq
                a0_ = sh_ahq[i][0]; a1_ = sh_ahq[i][1];
                a2_ = sh_ahq[i][2]; a3_ = sh_ahq[i][3];
            }
            const float c0 = __cosf(a0_ + sh_vqcw[g][0]);
            const float c1 = __cosf(a1_ + sh_vqcw[g][1]);
            const float c2 = __cosf(a2_ + sh_vqcw[g][2]);
            const float c3 = __cosf(a3_ + sh_vqcw[g][3]);
            sh_E[g][i][0] = c1 * c2 * c3;
            const float e1 = c0 * c1;
            sh_E[g][i][1] = e1;
            const float e2 = e1 * c2;
            sh_E[g][i][2] = e2;
            sh_E[g][i][3] = e2 * c3;
        }
        __syncthreads();

        // ---- Phase D: back-projection via V_WMMA_F32_16X16X4_F32 + GRU gating ----
        // A = E_g (16x4): lane holds E[nl][2*half .. 2*half+1]
        v2f a0, a1, a2, a3;
        a0.x = sh_E[0][nl][2*half]; a0.y = sh_E[0][nl][2*half + 1];
        a1.x = sh_E[1][nl][2*half]; a1.y = sh_E[1][nl][2*half + 1];
        a2.x = sh_E[2][nl][2*half]; a2.y = sh_E[2][nl][2*half + 1];
        a3.x = sh_E[3][nl][2*half]; a3.y = sh_E[3][nl][2*half + 1];
        const v8f cinit = {bn, bn, bn, bn, bn, bn, bn, bn};

        v8f dr  = __builtin_amdgcn_wmma_f32_16x16x4_f32(false, a0, false, bmat, (short)0, cinit, false, false);
        v8f dz  = __builtin_amdgcn_wmma_f32_16x16x4_f32(false, a1, false, bmat, (short)0, cinit, false, false);
        v8f dnx = __builtin_amdgcn_wmma_f32_16x16x4_f32(false, a2, false, bmat, (short)0, cinit, false, false);
        v8f dnh = __builtin_amdgcn_wmma_f32_16x16x4_f32(false, a3, false, bmat, (short)0, cinit, false, false);

        #pragma unroll
        for (int v = 0; v < 8; ++v) {
            const int m = v + 8 * half;             // C/D layout: M = vgpr + 8*half
            const float hold = sh_h[m][col];
            const float r  = fast_sigmoid(dr[v]);
            const float z  = fast_sigmoid(dz[v]);
            const float nt = fast_tanh(dnx[v] + r * dnh[v]);
            const float hn = (1.0f - z) * nt + z * hold;
            sh_h[m][col] = hn;
            out[((size_t)(b0 + m) * T_LEN + t) * H_DIM + col] = hn;
        }

        // commit prefetched x_{t+1}
        if (t + 1 < T_LEN) {
            float4* dst = reinterpret_cast<float4*>(&sh_x[nxt][row][sub * 8]);
            dst[0] = r0; dst[1] = r1;
        }
        __syncthreads();
    }

    // ---- h_last and c_last (zeros) ----
    const size_t baseL = (size_t)B_TOT * T_LEN * H_DIM;
    for (int k = tid; k < TILE * H_DIM; k += 256) {
        const int i = k >> 7, n = k & 127;
        out[baseL + (size_t)(b0 + i) * H_DIM + n] = sh_h[i][n];
        out[baseL + (size_t)B_TOT * H_DIM + (size_t)(b0 + i) * H_DIM + n] = 0.0f;
    }
}

extern "C" void kernel_launch(void* const* d_in, const int* in_sizes, int n_in,
                              void* d_out, int out_size, void* d_ws, size_t ws_size,
                              hipStream_t stream) {
    (void)in_sizes; (void)n_in; (void)out_size; (void)d_ws; (void)ws_size;
    const float* x     = (const float*)d_in[0];
    const float* W_in  = (const float*)d_in[1];
    const float* b_in  = (const float*)d_in[2];
    const float* W_x   = (const float*)d_in[3];
    const float* b_x   = (const float*)d_in[4];
    const float* W_h   = (const float*)d_in[5];
    const float* b_h   = (const float*)d_in[6];
    const float* W_out = (const float*)d_in[7];
    const float* b_out = (const float*)d_in[8];
    const float* vqc_w = (const float*)d_in[9];
    float* out = (float*)d_out;

    dim3 grid(B_TOT / TILE);   // 8 workgroups of 16 batch rows
    dim3 block(256);           // 8 wave32
    hipLaunchKernelGGL(qgru_kernel, grid, block, 0, stream,
                       x, W_in, b_in, W_x, b_x, W_h, b_h, W_out, b_out, vqc_w, out);
}